// SetCriterion_84945863180752
// MI455X (gfx1250) — compile-verified
//
#include <hip/hip_runtime.h>

// MI455X / gfx1250, wave32. Pairwise squared distances via V_WMMA_F32_16X16X4_F32:
//   d2  = (1, cy, cx, cy^2+cx^2) . (py^2+px^2, -2py, -2px, 1)
//   dy2 = (1, cy, cy^2, 0)       . (py^2,      -2py,  1,   0)
// Coordinates are integers <= 511 so all products < 2^24: fp32 WMMA accumulation
// is exact, making the Chebyshev truncation mask (dy2<=2025 && dx2<=2025) exact.

typedef __attribute__((ext_vector_type(2))) float v2f;
typedef __attribute__((ext_vector_type(8))) float v8f;

#define NB   8
#define NP   512
#define NQ   4096
#define IMGF 512.0f
#define D2MAX 2025.5f                         // 45^2 + tolerance (values are exact ints)
#define NEGINV (-1.0f / 450.0f)               // -1/(2*sigma^2)

// ---------------------------------------------------------------------------
// Density + bilinear + focal kernel. grid = (16, 8), block = 256 (8 waves).
// Wave w of block (bx, b) owns M-tile mtile = bx*8 + w (16 corners = 4 samples)
// of batch b, and loops over all 32 N-tiles of that batch's 512 points.
// ---------------------------------------------------------------------------
__global__ __launch_bounds__(256) void density_focal_kernel(
    const float* __restrict__ tp,     // target_points [NB*NP, 2]
    const float* __restrict__ sp,     // src_points    [NB*NP, 2]
    float* __restrict__ acc)          // acc[0] += sum of focal raw terms
{
    __shared__ float4 ldsP[NP];       // per point: {py^2+px^2, -2py, -2px, py^2}
    const int b   = blockIdx.y;
    const int tid = threadIdx.x;

    // Stage point descriptors for this batch into LDS (exact integer coords).
    for (int j = tid; j < NP; j += 256) {
        float px = fminf(fmaxf(rintf(tp[(b * NP + j) * 2 + 0]), 0.0f), 511.0f);
        float py = fminf(fmaxf(rintf(tp[(b * NP + j) * 2 + 1]), 0.0f), 511.0f);
        ldsP[j] = make_float4(py * py + px * px, -2.0f * py, -2.0f * px, py * py);
    }
    __syncthreads();

    const int lane  = tid & 31;
    const int wave  = tid >> 5;
    const int mtile = blockIdx.x * 8 + wave;   // 0..127
    const int m     = lane & 15;               // row within tile (both lane halves)
    const int c     = mtile * 16 + m;          // corner index within batch
    const int s     = c >> 2;                  // sample index within batch
    const int k     = c & 3;                   // 0:(x0,y0) 1:(x1,y0) 2:(x0,y1) 3:(x1,y1)

    // Corner coordinates (integers) from the predicted sample point.
    float sx = sp[(b * NP + s) * 2 + 0], sy = sp[(b * NP + s) * 2 + 1];
    float fx = fminf(fmaxf(sx * IMGF, 0.0f), 511.0f);
    float fy = fminf(fmaxf(sy * IMGF, 0.0f), 511.0f);
    float x0 = floorf(fx), y0 = floorf(fy);
    float x1 = fminf(x0 + 1.0f, 511.0f), y1 = fminf(y0 + 1.0f, 511.0f);
    float cx = (k & 1) ? x1 : x0;
    float cy = (k & 2) ? y1 : y0;

    const bool lo = lane < 16;   // A layout: lanes 0-15 hold K0,K1; lanes 16-31 hold K2,K3
    v2f aD; aD.x = lo ? 1.0f : cx;        aD.y = lo ? cy : (cy * cy + cx * cx);
    v2f aY; aY.x = lo ? 1.0f : cy * cy;   aY.y = lo ? cy : 0.0f;

    v8f mx = {};                 // running max of gaussian responses (>= 0)

    for (int t = 0; t < 32; ++t) {
        float4 pd = ldsP[t * 16 + (lane & 15)];
        v2f bD; bD.x = lo ? pd.x : pd.z;  bD.y = lo ? pd.y : 1.0f;
        v2f bY; bY.x = lo ? pd.w : 1.0f;  bY.y = lo ? pd.y : 0.0f;
        v8f z = {};
        v8f d2  = __builtin_amdgcn_wmma_f32_16x16x4_f32(false, aD, false, bD,
                                                        (short)0, z, false, false);
        v8f dy2 = __builtin_amdgcn_wmma_f32_16x16x4_f32(false, aY, false, bY,
                                                        (short)0, z, false, false);
#pragma unroll
        for (int kk = 0; kk < 8; ++kk) {
            float e2 = d2[kk];
            float ey = dy2[kk];
            float ex = e2 - ey;
            float g  = (ey <= D2MAX && ex <= D2MAX) ? __expf(e2 * NEGINV) : 0.0f;
            mx[kk] = fmaxf(mx[kk], g);
        }
    }

    // Max over the 16 lanes of each half (reduces over point index N).
#pragma unroll
    for (int off = 1; off < 16; off <<= 1) {
#pragma unroll
        for (int kk = 0; kk < 8; ++kk)
            mx[kk] = fmaxf(mx[kk], __shfl_xor(mx[kk], off, 32));
    }
    // Now: lanes 0-15 hold densities of corners 0..7 in mx[0..7];
    //      lanes 16-31 hold corners 8..15. Lanes 0 and 16 finish 2 samples each.
    float fsum = 0.0f;
    if ((lane & 15) == 0) {
#pragma unroll
        for (int t = 0; t < 2; ++t) {
            int sl = (lane >> 4) * 2 + t;          // local sample 0..3
            int sg = b * NP + mtile * 4 + sl;      // global sample
            float gsx = sp[sg * 2 + 0], gsy = sp[sg * 2 + 1];
            float gx = fminf(fmaxf(gsx * IMGF, 0.0f), 511.0f);
            float gy = fminf(fmaxf(gsy * IMGF, 0.0f), 511.0f);
            float wx = gx - floorf(gx), wy = gy - floorf(gy);
            float v00 = mx[4 * t + 0], v01 = mx[4 * t + 1];
            float v10 = mx[4 * t + 2], v11 = mx[4 * t + 3];
            float p = v00 * (1.0f - wx) * (1.0f - wy) + v01 * wx * (1.0f - wy)
                    + v10 * (1.0f - wx) * wy          + v11 * wx * wy;
            float om = 1.0f - p;
            fsum += -om * om * __logf(fmaxf(p, 1e-6f));
        }
    }
    fsum += __shfl_xor(fsum, 16, 32);
    if (lane == 0) atomicAdd(acc + 0, fsum);
}

// ---------------------------------------------------------------------------
// Workspace init: acc[0..3] = 0, flags[NB*NQ] = 0.
// ---------------------------------------------------------------------------
__global__ void zero_ws_kernel(float* __restrict__ acc, unsigned char* __restrict__ flags)
{
    int i = blockIdx.x * 256 + threadIdx.x;
    if (i < 4) acc[i] = 0.0f;
    if (i < NB * NQ) flags[i] = 0;
}

__global__ void scatter_kernel(const int* __restrict__ src_idx,
                               unsigned char* __restrict__ flags)
{
    int i = blockIdx.x * 256 + threadIdx.x;   // 0..NB*NP-1
    if (i < NB * NP) {
        int b = i >> 9;                        // P = 512
        flags[b * NQ + src_idx[i]] = 1;        // duplicates benign (idempotent)
    }
}

__global__ void ce_kernel(const float* __restrict__ logits,
                          const unsigned char* __restrict__ flags,
                          float* __restrict__ acc)
{
    int i = blockIdx.x * 256 + threadIdx.x;   // 0..NB*NQ-1
    float wnll = 0.0f, w = 0.0f;
    if (i < NB * NQ) {
        int t = flags[i];
        float l0 = logits[2 * i], l1 = logits[2 * i + 1];
        float mm = fmaxf(l0, l1);
        float lse = mm + __logf(__expf(l0 - mm) + __expf(l1 - mm));
        float lt = t ? l1 : l0;
        w = t ? 1.0f : 0.5f;
        wnll = w * (lse - lt);
    }
#pragma unroll
    for (int off = 1; off < 32; off <<= 1) {
        wnll += __shfl_xor(wnll, off, 32);
        w    += __shfl_xor(w, off, 32);
    }
    if ((threadIdx.x & 31) == 0) {
        atomicAdd(acc + 1, wnll);
        atomicAdd(acc + 2, w);
    }
}

__global__ void finalize_kernel(const float* __restrict__ acc, float* __restrict__ out)
{
    // loss = loss_ce + 5 * (0.1 * sum_raw / N) = wnll/w + 0.5*sum_raw/4096
    out[0] = acc[1] / acc[2] + 0.5f * acc[0] / 4096.0f;
}

// ---------------------------------------------------------------------------
extern "C" void kernel_launch(void* const* d_in, const int* in_sizes, int n_in,
                              void* d_out, int out_size, void* d_ws, size_t ws_size,
                              hipStream_t stream)
{
    const float* target_points = (const float*)d_in[0];   // [NB*NP, 2]
    const float* src_points    = (const float*)d_in[1];   // [NB*NP, 2]
    const float* pred_logits   = (const float*)d_in[2];   // [NB, NQ, 2]
    const int*   src_idx       = (const int*)d_in[4];     // [NB, NP]
    float* out = (float*)d_out;

    float* acc = (float*)d_ws;                             // 4 floats
    unsigned char* flags = (unsigned char*)d_ws + 16;      // NB*NQ bytes

    zero_ws_kernel<<<(NB * NQ + 255) / 256, 256, 0, stream>>>(acc, flags);
    scatter_kernel<<<(NB * NP + 255) / 256, 256, 0, stream>>>(src_idx, flags);
    ce_kernel<<<(NB * NQ + 255) / 256, 256, 0, stream>>>(pred_logits, flags, acc);
    density_focal_kernel<<<dim3(16, NB), 256, 0, stream>>>(target_points, src_points, acc);
    finalize_kernel<<<1, 1, 0, stream>>>(acc, out);

    (void)in_sizes; (void)n_in; (void)out_size; (void)ws_size;
}